// TemporalDGMRF_53893249630426
// MI455X (gfx1250) — compile-verified
//
#include <hip/hip_runtime.h>
#include <stdint.h>

#define BATCH 16     // B, fixed by reference
#define EPB   256    // edges per block == blockDim.x

// ---- tiled transpose: x[B,N] -> h[N,B], both phases coalesced ----
__global__ __launch_bounds__(256) void transpose_in_kernel(const float* __restrict__ x,
                                                           float* __restrict__ h, int N) {
  __shared__ float tile[16][17];           // +1 pad: no LDS bank conflicts
  const int n0 = blockIdx.x * 16;
  const int a  = threadIdx.x >> 4;         // 0..15
  const int c  = threadIdx.x & 15;         // 0..15
  // read phase: a = batch row, c = n offset -> consecutive lanes consecutive addrs
  const int nr = n0 + c;
  float v = 0.0f;
  if (nr < N) v = x[(size_t)a * N + nr];
  tile[c][a] = v;
  __syncthreads();
  // write phase: a = n offset, c = batch -> consecutive lanes consecutive addrs
  const int nw = n0 + a;
  if (nw < N) h[(nw << 4) + c] = tile[a][c];
}

// ---- tiled transpose: h[N,B] -> out[B,N], both phases coalesced ----
__global__ __launch_bounds__(256) void transpose_out_kernel(const float* __restrict__ h,
                                                            float* __restrict__ out, int N) {
  __shared__ float tile[16][17];
  const int n0 = blockIdx.x * 16;
  const int a  = threadIdx.x >> 4;
  const int c  = threadIdx.x & 15;
  // read phase: a = n offset, c = batch -> consecutive addrs over c
  const int nr = n0 + a;
  float v = 0.0f;
  if (nr < N) v = h[(nr << 4) + c];
  tile[a][c] = v;
  __syncthreads();
  // write phase: a = batch, c = n offset -> consecutive addrs over c
  const int nw = n0 + c;
  if (nw < N) out[(size_t)a * N + nw] = tile[c][a];
}

// ---- per-edge coefficients, computed once (constant over T steps) ----
__global__ __launch_bounds__(256) void coef_kernel(const float* __restrict__ ea,
                                                   const float* __restrict__ vel,
                                                   const float* __restrict__ dpar,
                                                   float* __restrict__ ci,
                                                   float* __restrict__ cj, int E) {
  int e = blockIdx.x * blockDim.x + threadIdx.x;
  if (e < E) {
    float v0 = vel[0], v1 = vel[1], d = dpar[0];
    float diff = d * d;
    float2 e2 = ((const float2*)ea)[e];                 // one b64 load
    float adv = -0.5f * (e2.x * v0 + e2.y * v1);
    ci[e] = adv - diff;
    cj[e] = adv + diff;
  }
}

// ---- one time step: hdst (pre-initialized to hsrc) += scatter_i(ci*h[i] + cj*h[j]) ----
// Edge tile staged via gfx1250 async global->LDS (ASYNCcnt). Full blocks use
// b128 per lane (4x fewer async ops); tail block falls back to clamped b32.
// 16 lanes cooperate per edge: every gather / atomic is one 64B cacheline.
__global__ __launch_bounds__(256) void edge_step_kernel(const int* __restrict__ ei,
                                                        const int* __restrict__ ej,
                                                        const float* __restrict__ ci,
                                                        const float* __restrict__ cj,
                                                        const float* __restrict__ hsrc,
                                                        float* __restrict__ hdst, int E) {
  __shared__ alignas(16) int   s_i[EPB];
  __shared__ alignas(16) int   s_j[EPB];
  __shared__ alignas(16) float s_ci[EPB];
  __shared__ alignas(16) float s_cj[EPB];

  const int t    = threadIdx.x;
  const int base = blockIdx.x * EPB;

  if (base + EPB <= E) {
    // full tile: 64 lanes per array, 16B per lane (wave-uniform array select)
    const int q    = t >> 6;          // 0..3 : which array
    const int r    = t & 63;          // slot group
    const int slot = r << 2;          // element index in tile
    const int goff = (base + slot) << 2;
    if (q == 0) {
      unsigned lo = (unsigned)(uintptr_t)(&s_i[slot]);
      asm volatile("global_load_async_to_lds_b128 %0, %1, %2"
                   :: "v"(lo), "v"(goff), "s"(ei) : "memory");
    } else if (q == 1) {
      unsigned lo = (unsigned)(uintptr_t)(&s_j[slot]);
      asm volatile("global_load_async_to_lds_b128 %0, %1, %2"
                   :: "v"(lo), "v"(goff), "s"(ej) : "memory");
    } else if (q == 2) {
      unsigned lo = (unsigned)(uintptr_t)(&s_ci[slot]);
      asm volatile("global_load_async_to_lds_b128 %0, %1, %2"
                   :: "v"(lo), "v"(goff), "s"(ci) : "memory");
    } else {
      unsigned lo = (unsigned)(uintptr_t)(&s_cj[slot]);
      asm volatile("global_load_async_to_lds_b128 %0, %1, %2"
                   :: "v"(lo), "v"(goff), "s"(cj) : "memory");
    }
  } else {
    // tail tile: clamped per-element b32 staging (compute loop guards e < E)
    int ge = base + t;
    if (ge >= E) ge = E - 1;
    const int goff = ge << 2;
    unsigned loff_i  = (unsigned)(uintptr_t)(&s_i[t]);
    unsigned loff_j  = (unsigned)(uintptr_t)(&s_j[t]);
    unsigned loff_ci = (unsigned)(uintptr_t)(&s_ci[t]);
    unsigned loff_cj = (unsigned)(uintptr_t)(&s_cj[t]);
    asm volatile("global_load_async_to_lds_b32 %0, %1, %2"
                 :: "v"(loff_i),  "v"(goff), "s"(ei) : "memory");
    asm volatile("global_load_async_to_lds_b32 %0, %1, %2"
                 :: "v"(loff_j),  "v"(goff), "s"(ej) : "memory");
    asm volatile("global_load_async_to_lds_b32 %0, %1, %2"
                 :: "v"(loff_ci), "v"(goff), "s"(ci) : "memory");
    asm volatile("global_load_async_to_lds_b32 %0, %1, %2"
                 :: "v"(loff_cj), "v"(goff), "s"(cj) : "memory");
  }
  asm volatile("s_wait_asynccnt 0" ::: "memory");
  __syncthreads();

  const int g = t >> 4;   // edge sub-group 0..15
  const int b = t & 15;   // batch lane 0..15
#pragma unroll
  for (int it = 0; it < 16; ++it) {
    const int el = g * 16 + it;
    const int e  = base + el;
    if (e < E) {
      const int   i  = s_i[el];
      const int   j  = s_j[el];
      const float hi = hsrc[(i << 4) + b];   // 64B cacheline, 16 coalesced lanes
      const float hj = hsrc[(j << 4) + b];
      const float m  = s_ci[el] * hi + s_cj[el] * hj;
      atomicAdd(&hdst[(i << 4) + b], m);     // global_atomic_add_f32 (no return)
    }
  }
}

extern "C" void kernel_launch(void* const* d_in, const int* in_sizes, int n_in,
                              void* d_out, int out_size, void* d_ws, size_t ws_size,
                              hipStream_t stream) {
  (void)n_in; (void)out_size; (void)ws_size;

  const int BN = in_sizes[0];        // B*N
  const int N  = BN / BATCH;
  const int E  = in_sizes[1] / 2;    // edge_index is [2,E]
  const int T  = 16;                 // T_STEPS compile-time constant in reference

  const float* x    = (const float*)d_in[0];
  const int*   eidx = (const int*)d_in[1];
  const float* ea   = (const float*)d_in[2];
  const float* vel  = (const float*)d_in[3];
  const float* dpar = (const float*)d_in[4];
  float* out = (float*)d_out;

  // Workspace: hA | hB | ci | cj  = 32MB + 32MB + 8MB + 8MB = 80MB
  float* hA = (float*)d_ws;
  float* hB = hA + (size_t)N * BATCH;
  float* ci = hB + (size_t)N * BATCH;
  float* cj = ci + (size_t)E;

  const int* ei = eidx;
  const int* ej = eidx + E;

  const int tpb = 256;
  transpose_in_kernel<<<(N + 15) / 16, tpb, 0, stream>>>(x, hA, N);
  coef_kernel<<<(E + tpb - 1) / tpb, tpb, 0, stream>>>(ea, vel, dpar, ci, cj, E);

  float* src = hA;
  float* dst = hB;
  for (int t = 0; t < T; ++t) {
    hipMemcpyAsync(dst, src, (size_t)N * BATCH * sizeof(float),
                   hipMemcpyDeviceToDevice, stream);
    edge_step_kernel<<<(E + EPB - 1) / EPB, tpb, 0, stream>>>(ei, ej, ci, cj, src, dst, E);
    float* tmp = src; src = dst; dst = tmp;
  }
  transpose_out_kernel<<<(N + 15) / 16, tpb, 0, stream>>>(src, out, N);
}